// CML_21861383537472
// MI455X (gfx1250) — compile-verified
//
#include <hip/hip_runtime.h>

#define N_ITEMS_C 100000
#define D_C 64
#define B_C 16384
#define NNEG_C 100
#define MARGIN_C 1.0f

typedef float v2f __attribute__((ext_vector_type(2)));
typedef float v8f __attribute__((ext_vector_type(8)));

__device__ __forceinline__ float wave_allreduce_add(float x) {
#pragma unroll
  for (int m = 16; m >= 1; m >>= 1) x += __shfl_xor(x, m, 32);
  return x;
}

// One wave32 per batch element. Computes metric[m] = margin + d_ij - ||u - w_m||^2
// for 100 negatives via V_WMMA_F32_16X16X4_F32 with an augmented K=68 vector:
//   A row m = [ w_m (64), ||w_m||^2, 1, 0, 0 ]
//   B col   = [ 2u  (64),     -1, margin + d_ij - ||u||^2, 0, 0 ]  (broadcast over N)
// The B operand is tile-invariant -> materialized ONCE in registers (bv[16]),
// so the 7-tile inner loop is pure global-gather + WMMA with no LDS traffic.
__global__ __launch_bounds__(256) void CML_warp_metric_kernel(
    const float* __restrict__ user_emb,
    const float* __restrict__ item_emb,
    const int*   __restrict__ idx_i,
    const int*   __restrict__ idx_j,
    const int*   __restrict__ idx_k,
    float*       __restrict__ per_b) {
  const int lane = threadIdx.x & 31;
  const int wave = threadIdx.x >> 5;
  const int b    = blockIdx.x * 8 + wave;
  const bool hi  = (lane >= 16);

  // ---- u / v lookup with max_norm=1 clipping; lane owns dims {2*lane, 2*lane+1}
  const int ui = idx_i[b];
  const int vj = idx_j[b];
  const float* urow = user_emb + (long long)ui * D_C;
  const float* vrow = item_emb + (long long)vj * D_C;
  float ux = urow[2 * lane], uy = urow[2 * lane + 1];
  float vx = vrow[2 * lane], vy = vrow[2 * lane + 1];

  float nu2 = wave_allreduce_add(ux * ux + uy * uy);
  float nv2 = wave_allreduce_add(vx * vx + vy * vy);
  float nu = sqrtf(nu2), nv = sqrtf(nv2);
  float su = (nu > 1.0f) ? (1.0f / (nu + 1e-7f)) : 1.0f;
  float sv = (nv > 1.0f) ? (1.0f / (nv + 1e-7f)) : 1.0f;
  ux *= su; uy *= su; vx *= sv; vy *= sv;

  float ddx = ux - vx, ddy = uy - vy;
  float d_ij = wave_allreduce_add(ddx * ddx + ddy * ddy);
  float usq  = nu2 * su * su;                 // ||u_clipped||^2
  float cterm = MARGIN_C + d_ij - usq;        // B[65]
  const float bu_x = 2.0f * ux, bu_y = 2.0f * uy;  // distributed 2u

  // ---- Hoisted B operand: bv[c] holds u dims {4c,4c+1} (lanes 0-15) or
  //      {4c+2,4c+3} (lanes 16-31), matching the f32 4x16 B-matrix layout.
  v2f bv[16];
#pragma unroll
  for (int c = 0; c < 16; ++c) {
    int src = 2 * c + (hi ? 1 : 0);
    bv[c].x = __shfl(bu_x, src, 32);
    bv[c].y = __shfl(bu_y, src, 32);
  }
  v2f bv_epi;                                  // K=64..67 of B
  bv_epi.x = hi ? 0.0f : -1.0f;
  bv_epi.y = hi ? 0.0f : cterm;

  float pos_sum = 0.0f, cnt = 0.0f;
  const int* krow = idx_k + (long long)b * NNEG_C;

  for (int t = 0; t < 7; ++t) {  // 7 tiles of 16 negatives (112, padded, masked)
    int n  = 16 * t + (lane & 15);
    int nc = (n < NNEG_C) ? n : (NNEG_C - 1);
    int widx = krow[nc];
    // f32 A-matrix 16x4 layout: lanes 0-15 hold K=4c,4c+1 ; lanes 16-31 hold K=4c+2,4c+3
    const float* wr = item_emb + (long long)widx * D_C + (hi ? 2 : 0);

    v2f wrow[16];
    float wsq = 0.0f;
#pragma unroll
    for (int c = 0; c < 16; ++c) {
      wrow[c] = *(const v2f*)(wr + 4 * c);    // strided 8B gathers, half-row per lane
      wsq += wrow[c].x * wrow[c].x + wrow[c].y * wrow[c].y;
    }
    wsq += __shfl_xor(wsq, 16, 32);           // full ||w_m||^2 in both half-lanes

    v8f acc = {};
#pragma unroll
    for (int c = 0; c < 16; ++c) {
      acc = __builtin_amdgcn_wmma_f32_16x16x4_f32(false, wrow[c], false, bv[c],
                                                  (short)0, acc, false, false);
    }
    {   // epilogue chunk K=64..67: A=[||w||^2, 1 | 0,0], B=[-1, cterm | 0,0]
      v2f a;
      a.x = hi ? 0.0f : wsq;
      a.y = hi ? 0.0f : 1.0f;
      acc = __builtin_amdgcn_wmma_f32_16x16x4_f32(false, a, false, bv_epi,
                                                  (short)0, acc, false, false);
    }
    // C/D layout: VGPR g, lanes 0-15 -> M=g, lanes 16-31 -> M=g+8; all N columns equal.
    if ((lane & 15) == 0) {
      int mbase = 16 * t + (hi ? 8 : 0);
#pragma unroll
      for (int g = 0; g < 8; ++g) {
        float mval = acc[g];
        if (mbase + g < NNEG_C) {
          pos_sum += fmaxf(mval, 0.0f);
          cnt     += (mval > 0.0f) ? 1.0f : 0.0f;
        }
      }
    }
  }
  pos_sum += __shfl_xor(pos_sum, 16, 32);
  cnt     += __shfl_xor(cnt, 16, 32);
  if (lane == 0) {
    float w_ij = logf((float)N_ITEMS_C * cnt / (float)NNEG_C + 1.0f);
    per_b[b] = w_ij * pos_sum;
  }
}

// Deterministic fixed-order reduction of 16384 partials -> scalar (no float atomics).
__global__ __launch_bounds__(256) void CML_reduce_kernel(const float* __restrict__ per_b,
                                                         float* __restrict__ out) {
  __shared__ float smem[256];
  float s = 0.0f;
  for (int i = threadIdx.x; i < B_C; i += 256) s += per_b[i];
  smem[threadIdx.x] = s;
  __syncthreads();
  for (int off = 128; off >= 1; off >>= 1) {
    if ((int)threadIdx.x < off) smem[threadIdx.x] += smem[threadIdx.x + off];
    __syncthreads();
  }
  if (threadIdx.x == 0) out[0] = smem[0];
}

extern "C" void kernel_launch(void* const* d_in, const int* in_sizes, int n_in,
                              void* d_out, int out_size, void* d_ws, size_t ws_size,
                              hipStream_t stream) {
  const float* user_emb = (const float*)d_in[0];
  const float* item_emb = (const float*)d_in[1];
  const int*   i_idx    = (const int*)d_in[2];
  const int*   j_idx    = (const int*)d_in[3];
  const int*   k_idx    = (const int*)d_in[4];
  float* per_b = (float*)d_ws;   // needs B_C * 4 = 64 KB of scratch

  CML_warp_metric_kernel<<<B_C / 8, 256, 0, stream>>>(user_emb, item_emb,
                                                      i_idx, j_idx, k_idx, per_b);
  CML_reduce_kernel<<<1, 256, 0, stream>>>(per_b, (float*)d_out);
}